// DistortionLoss_45200235823192
// MI455X (gfx1250) — compile-verified
//
#include <hip/hip_runtime.h>
#include <hip/hip_bf16.h>
#include <math.h>

typedef __attribute__((ext_vector_type(2))) float v2f;
typedef __attribute__((ext_vector_type(8))) float v8f;

#define EPSV 1e-8f

// ---------------------------------------------------------------------------
// Zero the 4 double accumulators in workspace.
__global__ void dl_init_kernel(double* acc) {
    if (threadIdx.x < 4) acc[threadIdx.x] = 0.0;
}

// ---------------------------------------------------------------------------
// Row squared norms: sqn[i] = sum_k M[i,k]^2.  One thread per row.
__global__ void dl_sqnorm_kernel(const float* __restrict__ M,
                                 float* __restrict__ sqn, int N, int K) {
    int i = blockIdx.x * blockDim.x + threadIdx.x;
    if (i >= N) return;
    const float4* row = (const float4*)(M + (size_t)i * K);
    float s = 0.f;
    for (int k = 0; k < (K >> 2); ++k) {
        float4 v = row[k];
        s += v.x * v.x + v.y * v.y + v.z * v.z + v.w * v.w;
    }
    sqn[i] = s;
}

// ---------------------------------------------------------------------------
// Fused pass: per 16x16 tile, compute Gram via V_WMMA_F32_16X16X4_F32,
// form d, a=d/denom, b=D/denom, accumulate S1..S4 globally.
// One wave (32 lanes) per tile; no divergence before WMMA (EXEC all ones).
__global__ void __launch_bounds__(256)
dl_main_kernel(const float* __restrict__ M, const float* __restrict__ Dm,
               const float* __restrict__ sqn, double* __restrict__ acc,
               int N, int K) {
    const int lane  = threadIdx.x & 31;
    const int wave  = threadIdx.x >> 5;
    const int wpb   = blockDim.x >> 5;               // waves per block (8)
    const int tilesPerRow = N >> 4;
    const int waveId = blockIdx.x * wpb + wave;
    const int tileI = waveId / tilesPerRow;
    const int tileJ = waveId % tilesPerRow;
    const int i0 = tileI << 4;
    const int j0 = tileJ << 4;

    const int lmod = lane & 15;   // row-within-tile (A) / col-within-tile (B,C)
    const int lhi  = lane >> 4;   // K-half selector for A/B frags

    // --- Preload D tile (streamed once -> non-temporal) + sq-norms early ---
    float dvals[8], sqni[8];
#pragma unroll
    for (int p = 0; p < 8; ++p) {
        int m = p + (lhi << 3);
        dvals[p] = __builtin_nontemporal_load(
                       &Dm[(size_t)(i0 + m) * N + (j0 + lmod)]);
        sqni[p]  = sqn[i0 + m];
    }
    float sqnj = sqn[j0 + lmod];

    // --- Gram tile: chained fp32 WMMA 16x16x4 over K ------------------------
    // A frag lane layout (16x4 f32): lanes 0-15 -> K=k,k+1 ; lanes 16-31 -> K=k+2,k+3
    // B frag (4x16 f32) for B = M^T has the identical per-lane gather, row base j0.
    v8f c = {};
    if (K == 64) {
        // Fast path: compile-time K. Batch all 32 frag loads (one base addr +
        // immediate offsets, no per-step waits), then run the 16-WMMA chain.
        const float* arow = M + ((size_t)(i0 + lmod) << 6) + (lhi << 1);
        const float* brow = M + ((size_t)(j0 + lmod) << 6) + (lhi << 1);
        v2f af[16], bf[16];
#pragma unroll
        for (int kf = 0; kf < 16; ++kf) {
            af[kf] = *(const v2f*)(arow + (kf << 2));
            bf[kf] = *(const v2f*)(brow + (kf << 2));
        }
#pragma unroll
        for (int kf = 0; kf < 16; ++kf) {
            c = __builtin_amdgcn_wmma_f32_16x16x4_f32(
                    /*neg_a=*/false, af[kf], /*neg_b=*/false, bf[kf],
                    /*c_mod=*/(short)0, c, /*reuse_a=*/false, /*reuse_b=*/false);
        }
    } else {
        const float* arow = M + (size_t)(i0 + lmod) * K + (lhi << 1);
        const float* brow = M + (size_t)(j0 + lmod) * K + (lhi << 1);
        for (int k = 0; k < K; k += 4) {
            v2f af = *(const v2f*)(arow + k);
            v2f bf = *(const v2f*)(brow + k);
            c = __builtin_amdgcn_wmma_f32_16x16x4_f32(
                    false, af, false, bf, (short)0, c, false, false);
        }
    }

    // --- Elementwise: C/D layout -> VGPR p holds (M=p+8*lhi, N=lmod) --------
    float s1 = 0.f, s2 = 0.f, s3 = 0.f, s4 = 0.f;
#pragma unroll
    for (int p = 0; p < 8; ++p) {
        int gi = i0 + p + (lhi << 3);
        int gj = j0 + lmod;
        float sq = sqni[p] + sqnj - 2.0f * c[p];
        float d  = (sq > 0.f) ? sqrtf(sq) : 0.f;
        float Dv = dvals[p];
        float denom = Dv + ((gi == gj) ? 1.0f : 0.0f) + EPSV;
        float inv = 1.0f / denom;
        float a = d * inv;
        float b = Dv * inv;
        s1 += a;
        s2 += a * a;
        s3 += a * b;
        s4 += b * b;
    }

    // --- wave32 reduction ---------------------------------------------------
#pragma unroll
    for (int off = 16; off > 0; off >>= 1) {
        s1 += __shfl_xor(s1, off, 32);
        s2 += __shfl_xor(s2, off, 32);
        s3 += __shfl_xor(s3, off, 32);
        s4 += __shfl_xor(s4, off, 32);
    }

    // --- block reduction in LDS, one f64 atomic per accumulator per block --
    __shared__ float red[8][4];
    if (lane == 0) {
        red[wave][0] = s1; red[wave][1] = s2;
        red[wave][2] = s3; red[wave][3] = s4;
    }
    __syncthreads();
    if (threadIdx.x == 0) {
        double t0 = 0, t1 = 0, t2 = 0, t3 = 0;
        for (int w = 0; w < wpb; ++w) {
            t0 += (double)red[w][0];
            t1 += (double)red[w][1];
            t2 += (double)red[w][2];
            t3 += (double)red[w][3];
        }
        atomicAdd(&acc[0], t0);
        atomicAdd(&acc[1], t1);
        atomicAdd(&acc[2], t2);
        atomicAdd(&acc[3], t3);
    }
}

// ---------------------------------------------------------------------------
// out = (scaling^2*S2 - 2*scaling*S3 + S4) / (N^2 - N), scaling = S1/S2
__global__ void dl_final_kernel(const double* __restrict__ acc,
                                float* __restrict__ out, int N) {
    if (threadIdx.x != 0) return;
    double S1 = acc[0], S2 = acc[1], S3 = acc[2], S4 = acc[3];
    double scaling = S1 / S2;
    double cnt = (double)N * (double)N - (double)N;
    double v = (scaling * scaling * S2 - 2.0 * scaling * S3 + S4) / cnt;
    out[0] = (float)v;
}

// ---------------------------------------------------------------------------
extern "C" void kernel_launch(void* const* d_in, const int* in_sizes, int n_in,
                              void* d_out, int out_size, void* d_ws, size_t ws_size,
                              hipStream_t stream) {
    const float* M  = (const float*)d_in[0];   // mapping [N, K] fp32
    const float* Dm = (const float*)d_in[1];   // D       [N, N] fp32
    float* out = (float*)d_out;

    // Derive shapes: in_sizes[1] = N*N, in_sizes[0] = N*K
    int N = (int)(sqrt((double)in_sizes[1]) + 0.5);
    int K = in_sizes[0] / N;                   // 64 for the reference

    // Workspace layout: [0,32B) 4 double accumulators; [256B, ...) sqn[N] f32
    double* acc = (double*)d_ws;
    float*  sqn = (float*)((char*)d_ws + 256);

    dl_init_kernel<<<1, 32, 0, stream>>>(acc);

    dl_sqnorm_kernel<<<(N + 255) / 256, 256, 0, stream>>>(M, sqn, N, K);

    int tilesPerRow = N / 16;
    int tiles = tilesPerRow * tilesPerRow;     // 65536
    int wavesPerBlock = 8;                     // 256 threads, wave32
    int blocks = tiles / wavesPerBlock;        // 8192
    dl_main_kernel<<<blocks, 256, 0, stream>>>(M, Dm, sqn, acc, N, K);

    dl_final_kernel<<<1, 32, 0, stream>>>(acc, out, N);
}